// PACA_42863773614891
// MI455X (gfx1250) — compile-verified
//
#include <hip/hip_runtime.h>
#include <hip/hip_bf16.h>
#include <math.h>

// ---------------- problem constants (match reference) ----------------
#define B_    4
#define KL_   128
#define XL_   4096
#define DIM_  256
#define H_    8
#define HD_   32
#define PB_   8
#define QKV_LD 768
#define SCALE_ 0.1767766952966369f   // 1/sqrt(32)

typedef __attribute__((ext_vector_type(2))) float v2f;
typedef __attribute__((ext_vector_type(8))) float v8f;

// D = A(16x4,f32) * B(4x16,f32) + C(16x16,f32)  -- CDNA5 V_WMMA_F32_16X16X4_F32
__device__ __forceinline__ v8f wmma4(v2f a, v2f b, v8f c) {
  return __builtin_amdgcn_wmma_f32_16x16x4_f32(false, a, false, b, (short)0, c,
                                               false, false);
}

// ---------------------------------------------------------------------
// Generic fp32 WMMA GEMM: C[M,N] = A[M,K] @ B[K,N] (+ bias[N]).
// One 16x16 output tile per wave32. All dims multiples of 16 (K mult of 4).
// ---------------------------------------------------------------------
__global__ void gemm_wmma_f32(const float* __restrict__ A, int lda,
                              const float* __restrict__ Bm, int ldb,
                              float* __restrict__ C, int ldc,
                              int M, int N, int K,
                              const float* __restrict__ bias) {
  const int wavesPerBlk = blockDim.x >> 5;
  const int wave = blockIdx.x * wavesPerBlk + (threadIdx.x >> 5);
  const int ntiles = N >> 4;
  const int total = (M >> 4) * ntiles;
  if (wave >= total) return;                 // wave-uniform
  const int mt = wave / ntiles, nt = wave - mt * ntiles;
  const int row0 = mt << 4, col0 = nt << 4;
  const int lane = threadIdx.x & 31;
  const int ln16 = lane & 15;                // A-row m / B-col n
  const int k0   = (lane >> 4) << 1;         // K sub-offset {0,2}
  const int mbase = (lane >> 4) << 3;        // C rows {0,8}

  v8f acc = {};
  const float* ap = A + (size_t)(row0 + ln16) * lda + k0;
  const float* bp = Bm + (size_t)k0 * ldb + col0 + ln16;
  for (int kk = 0; kk < K; kk += 4) {
    v2f a, b;
    a.x = ap[kk];       a.y = ap[kk + 1];
    b.x = bp[(size_t)kk * ldb];
    b.y = bp[(size_t)(kk + 1) * ldb];
    acc = wmma4(a, b, acc);
  }
  const float bs = bias ? bias[col0 + ln16] : 0.0f;
#pragma unroll
  for (int i = 0; i < 8; ++i)
    C[(size_t)(row0 + mbase + i) * ldc + col0 + ln16] = acc[i] + bs;
}

// ---------------------------------------------------------------------
// kernel->x attention. One workgroup (8 waves, 256 thr) per (b, h, k-tile16).
// Holds the full 16 x 4096 logit strip in LDS (256KB of CDNA5's 320KB/WGP).
// Also computes main_ori (polar-bin argmax of |scores|) and publishes it.
// ---------------------------------------------------------------------
__global__ void attn_kx_kernel(const float* __restrict__ qkv_x,
                               const float* __restrict__ qkv_k,
                               const int* __restrict__ rd,
                               const int* __restrict__ polar_pos,
                               const unsigned char* __restrict__ att_mask,
                               const float* __restrict__ dis_embed,
                               const float* __restrict__ polar_emb,
                               int* __restrict__ main_ori,
                               float* __restrict__ k_ctx) {
  const int bid = blockIdx.x;
  const int kt = bid & 7;            // KL/16 = 8
  const int h  = (bid >> 3) & 7;
  const int b  = bid >> 6;

  extern __shared__ float smem[];
  float* P      = smem;              // 16*4096 score/prob strip
  float* red    = P + 16 * XL_;      // 256   per-row partials
  float* rowmax = red + 256;         // 16
  float* rowsum = rowmax + 16;       // 16
  float* obins  = rowsum + 16;       // 256*8 orientation bins
  float* part   = obins + 2048;      // 8 waves * 16*32 ctx partials
  int*   ori_l  = (int*)(part + 4096);  // 16

  const int tid = threadIdx.x;
  const int wv = tid >> 5, lane = tid & 31;
  const int ln16 = lane & 15;
  const int k0 = (lane >> 4) << 1;
  const int mbase = (lane >> 4) << 3;

  // ---- stage 1: raw scaled scores S = k_q . k^T * scale  (WMMA) ----
  const float* Abase =
      qkv_k + (size_t)(b * KL_ + kt * 16 + ln16) * QKV_LD + h * HD_;  // k_q
  for (int xt = wv; xt < (XL_ >> 4); xt += 8) {
    const int x0 = xt << 4;
    const float* Bbase =
        qkv_x + (size_t)(b * XL_ + x0 + ln16) * QKV_LD + DIM_ + h * HD_;  // k
    v8f acc = {};
#pragma unroll
    for (int kk = 0; kk < HD_; kk += 4) {
      v2f a, bb;
      a.x  = Abase[kk + k0]; a.y  = Abase[kk + k0 + 1];
      bb.x = Bbase[kk + k0]; bb.y = Bbase[kk + k0 + 1];
      acc = wmma4(a, bb, acc);
    }
#pragma unroll
    for (int i = 0; i < 8; ++i)
      P[(mbase + i) * XL_ + x0 + ln16] = acc[i] * SCALE_;
  }
  __syncthreads();

  // ---- stage 2: orientation bins + argmax -> main_ori ----
  const int row = tid >> 4, j = tid & 15;      // 16 workers per k-row
  const int* pprow = polar_pos + ((size_t)b * KL_ + kt * 16 + row) * XL_;
#pragma unroll
  for (int o = 0; o < 8; ++o) obins[tid * 8 + o] = 0.0f;
  for (int x = j; x < XL_; x += 16)
    obins[tid * 8 + (pprow[x] & 7)] += fabsf(P[row * XL_ + x]);
  __syncthreads();
  if (tid < 16) {
    float best = -1.0f; int bi = 0;
    for (int o = 0; o < 8; ++o) {
      float s = 0.0f;
      for (int j2 = 0; j2 < 16; ++j2) s += obins[(tid * 16 + j2) * 8 + o];
      if (s > best) { best = s; bi = o; }      // first max, like jnp.argmax
    }
    ori_l[tid] = bi;
    main_ori[(b * H_ + h) * KL_ + kt * 16 + tid] = bi;
  }
  __syncthreads();

  // ---- stage 3: mask + rd_e + polar bias, softmax over x ----
  {
    const int* rdrow = rd + ((size_t)b * KL_ + kt * 16 + row) * XL_;
    const unsigned char* mrow =
        att_mask + ((size_t)(b * H_ + h) * XL_) * KL_ + (kt * 16 + row);
    const int ori = ori_l[row];
    float mx = -INFINITY;
    for (int x = j; x < XL_; x += 16) {
      float s = P[row * XL_ + x];
      if (mrow[(size_t)x * KL_]) s = -1.0e6f;  // mask BEFORE biases (ref order)
      s += dis_embed[rdrow[x] * H_ + h];
      s += polar_emb[(pprow[x] - ori + PB_) & 7];
      P[row * XL_ + x] = s;
      mx = fmaxf(mx, s);
    }
    red[row * 16 + j] = mx;
    __syncthreads();
    if (j == 0) {
      float m2 = red[row * 16];
      for (int t = 1; t < 16; ++t) m2 = fmaxf(m2, red[row * 16 + t]);
      rowmax[row] = m2;
    }
    __syncthreads();
    const float rm = rowmax[row];
    float sm = 0.0f;
    for (int x = j; x < XL_; x += 16) {
      float e = expf(P[row * XL_ + x] - rm);
      P[row * XL_ + x] = e;
      sm += e;
    }
    red[row * 16 + j] = sm;
    __syncthreads();
    if (j == 0) {
      float s2 = 0.0f;
      for (int t = 0; t < 16; ++t) s2 += red[row * 16 + t];
      rowsum[row] = s2;
    }
    __syncthreads();
  }

  // ---- stage 4: ctx = P @ V  (WMMA, x split across 8 waves) ----
#pragma unroll
  for (int dt = 0; dt < 2; ++dt) {
    v8f acc = {};
    const float* Bb = qkv_x + (size_t)(b * XL_) * QKV_LD + 2 * DIM_ +
                      h * HD_ + dt * 16 + ln16;   // v
    for (int xk = wv * 512; xk < wv * 512 + 512; xk += 4) {
      v2f a, bb;
      a.x  = P[ln16 * XL_ + xk + k0];
      a.y  = P[ln16 * XL_ + xk + k0 + 1];
      bb.x = Bb[(size_t)(xk + k0) * QKV_LD];
      bb.y = Bb[(size_t)(xk + k0 + 1) * QKV_LD];
      acc = wmma4(a, bb, acc);
    }
#pragma unroll
    for (int i = 0; i < 8; ++i)
      part[wv * 512 + (mbase + i) * 32 + dt * 16 + ln16] = acc[i];
  }
  __syncthreads();
  for (int e = tid; e < 512; e += 256) {
    const int r = e >> 5, d = e & 31;
    float s = 0.0f;
#pragma unroll
    for (int w = 0; w < 8; ++w) s += part[w * 512 + r * 32 + d];
    s /= rowsum[r];
    k_ctx[(size_t)(b * KL_ + kt * 16 + r) * DIM_ + h * HD_ + d] = s;
  }
}

// ---------------------------------------------------------------------
// x->kernel attention. One workgroup (8 waves) per (b, h, x-tile16).
// 16 x 128 logits in LDS; WMMA for scores and context.
// ---------------------------------------------------------------------
__global__ void attn_xk_kernel(const float* __restrict__ qkv_x,
                               const float* __restrict__ qkv_k,
                               const int* __restrict__ rd,
                               const int* __restrict__ polar_pos,
                               const unsigned char* __restrict__ att_mask,
                               const float* __restrict__ dis_embed,
                               const float* __restrict__ polar_emb,
                               const int* __restrict__ main_ori,
                               float* __restrict__ x_ctx) {
  const int bid = blockIdx.x;
  const int xt = bid & 255;          // XL/16
  const int h  = (bid >> 8) & 7;
  const int b  = bid >> 11;

  __shared__ float L[16 * KL_];
  __shared__ float red[256];
  __shared__ float rowmax[16], rowsum[16];

  const int tid = threadIdx.x;
  const int wv = tid >> 5, lane = tid & 31;
  const int ln16 = lane & 15;
  const int k0 = (lane >> 4) << 1;
  const int mbase = (lane >> 4) << 3;

  // ---- scores for k-column tile wv: S = q . k_k^T * scale ----
  {
    const float* Ab =
        qkv_x + (size_t)(b * XL_ + xt * 16 + ln16) * QKV_LD + h * HD_;     // q
    const float* Bb =
        qkv_k + (size_t)(b * KL_ + wv * 16 + ln16) * QKV_LD + DIM_ + h * HD_; // k_k
    v8f acc = {};
#pragma unroll
    for (int kk = 0; kk < HD_; kk += 4) {
      v2f a, bb;
      a.x  = Ab[kk + k0]; a.y  = Ab[kk + k0 + 1];
      bb.x = Bb[kk + k0]; bb.y = Bb[kk + k0 + 1];
      acc = wmma4(a, bb, acc);
    }
#pragma unroll
    for (int i = 0; i < 8; ++i) {
      const int xr = xt * 16 + mbase + i;   // x row (global in [0,XL))
      const int kc = wv * 16 + ln16;        // k col
      float s = acc[i] * SCALE_;
      if (att_mask[((size_t)(b * H_ + h) * XL_ + xr) * KL_ + kc]) s = -1.0e9f;
      s += dis_embed[rd[((size_t)b * KL_ + kc) * XL_ + xr] * H_ + h];
      const int ori = main_ori[(b * H_ + h) * KL_ + kc];
      s += polar_emb[(polar_pos[((size_t)b * KL_ + kc) * XL_ + xr] - ori + PB_) & 7];
      L[(mbase + i) * KL_ + kc] = s;
    }
  }
  __syncthreads();

  // ---- softmax over kl (128) ----
  const int row = tid >> 4, j = tid & 15;
  float mx = -INFINITY;
  for (int k = j; k < KL_; k += 16) mx = fmaxf(mx, L[row * KL_ + k]);
  red[row * 16 + j] = mx;
  __syncthreads();
  if (j == 0) {
    float m2 = red[row * 16];
    for (int t = 1; t < 16; ++t) m2 = fmaxf(m2, red[row * 16 + t]);
    rowmax[row] = m2;
  }
  __syncthreads();
  const float rm = rowmax[row];
  float sm = 0.0f;
  for (int k = j; k < KL_; k += 16) {
    float e = expf(L[row * KL_ + k] - rm);
    L[row * KL_ + k] = e;
    sm += e;
  }
  red[row * 16 + j] = sm;
  __syncthreads();
  if (j == 0) {
    float s2 = 0.0f;
    for (int t = 0; t < 16; ++t) s2 += red[row * 16 + t];
    rowsum[row] = s2;
  }
  __syncthreads();

  // ---- ctx = P(16x128) @ k_v(128x32); waves 0,1 take d-tiles 0,1 ----
  if (wv < 2) {                       // wave-uniform branch (EXEC full inside)
    const int dt = wv;
    v8f acc = {};
    const float* Bb = qkv_k + (size_t)(b * KL_) * QKV_LD + 2 * DIM_ +
                      h * HD_ + dt * 16 + ln16;   // k_v
#pragma unroll
    for (int kk = 0; kk < KL_; kk += 4) {
      v2f a, bb;
      a.x  = L[ln16 * KL_ + kk + k0];
      a.y  = L[ln16 * KL_ + kk + k0 + 1];
      bb.x = Bb[(size_t)(kk + k0) * QKV_LD];
      bb.y = Bb[(size_t)(kk + k0 + 1) * QKV_LD];
      acc = wmma4(a, bb, acc);
    }
#pragma unroll
    for (int i = 0; i < 8; ++i) {
      const int r = mbase + i;
      x_ctx[(size_t)(b * XL_ + xt * 16 + r) * DIM_ + h * HD_ + dt * 16 + ln16] =
          acc[i] / rowsum[r];
    }
  }
}

// ---------------------------------------------------------------------
extern "C" void kernel_launch(void* const* d_in, const int* in_sizes, int n_in,
                              void* d_out, int out_size, void* d_ws,
                              size_t ws_size, hipStream_t stream) {
  (void)in_sizes; (void)n_in; (void)out_size; (void)ws_size;
  const float* x          = (const float*)d_in[0];
  const float* kernal     = (const float*)d_in[1];
  const int*   rd         = (const int*)d_in[2];
  const int*   polar_pos  = (const int*)d_in[3];
  const unsigned char* am = (const unsigned char*)d_in[4];  // bool mask
  const float* W_qkv      = (const float*)d_in[5];
  const float* dis_embed  = (const float*)d_in[6];
  const float* polar_emb  = (const float*)d_in[7];
  const float* W_proj     = (const float*)d_in[8];
  const float* b_proj     = (const float*)d_in[9];
  float* out = (float*)d_out;                 // x_out then k_out

  // workspace carve (~69.2 MB)
  float* qkv_x = (float*)d_ws;                        // 16384*768
  float* qkv_k = qkv_x + (size_t)B_ * XL_ * QKV_LD;   // 512*768
  float* x_ctx = qkv_k + (size_t)B_ * KL_ * QKV_LD;   // 16384*256
  float* k_ctx = x_ctx + (size_t)B_ * XL_ * DIM_;     // 512*256
  int* main_ori = (int*)(k_ctx + (size_t)B_ * KL_ * DIM_);  // 4096

  const int MX = B_ * XL_;   // 16384
  const int MK = B_ * KL_;   // 512

  // 1) qkv_x = x @ W_qkv
  gemm_wmma_f32<<<(MX / 16) * (QKV_LD / 16) / 4, 128, 0, stream>>>(
      x, DIM_, W_qkv, QKV_LD, qkv_x, QKV_LD, MX, QKV_LD, DIM_, nullptr);
  // 2) qkv_k = kernal @ W_qkv
  gemm_wmma_f32<<<(MK / 16) * (QKV_LD / 16) / 4, 128, 0, stream>>>(
      kernal, DIM_, W_qkv, QKV_LD, qkv_k, QKV_LD, MK, QKV_LD, DIM_, nullptr);

  // 3) kernel->x attention (+ main_ori, k_ctx). 256 blocks, 288KB dyn LDS.
  const size_t ldsA =
      (size_t)(16 * XL_ + 256 + 16 + 16 + 2048 + 4096 + 16) * 4;
  attn_kx_kernel<<<B_ * H_ * (KL_ / 16), 256, ldsA, stream>>>(
      qkv_x, qkv_k, rd, polar_pos, am, dis_embed, polar_emb, main_ori, k_ctx);

  // 4) x->kernel attention (x_ctx)
  attn_xk_kernel<<<B_ * H_ * (XL_ / 16), 256, 0, stream>>>(
      qkv_x, qkv_k, rd, polar_pos, am, dis_embed, polar_emb, main_ori, x_ctx);

  // 5) x_out = x_ctx @ W_proj + b_proj
  gemm_wmma_f32<<<(MX / 16) * (DIM_ / 16) / 4, 128, 0, stream>>>(
      x_ctx, DIM_, W_proj, DIM_, out, DIM_, MX, DIM_, DIM_, b_proj);
  // 6) k_out = k_ctx @ W_proj + b_proj  (after x_out in flat output)
  gemm_wmma_f32<<<(MK / 16) * (DIM_ / 16) / 4, 128, 0, stream>>>(
      k_ctx, DIM_, W_proj, DIM_, out + (size_t)MX * DIM_, DIM_, MK, DIM_,
      DIM_, b_proj);
}